// GINNet_34067680592554
// MI455X (gfx1250) — compile-verified
//
#include <hip/hip_runtime.h>

#define N_NODES 10000
#define N_EDGES 320000
#define DIM     256

typedef __attribute__((ext_vector_type(2))) float v2f;
typedef __attribute__((ext_vector_type(8))) float v8f;

// ---------------------------------------------------------------------------
// Kernel 1: h_in = (1 + eps) * x    (seed the GIN accumulator with self term)
// ---------------------------------------------------------------------------
__global__ void init_h(const float* __restrict__ x, const float* __restrict__ eps,
                       float* __restrict__ h) {
    int gid = blockIdx.x * blockDim.x + threadIdx.x;       // float4 chunks
    if (gid < N_NODES * (DIM / 4)) {
        float s = 1.0f + eps[0];
        float4 v = ((const float4*)x)[gid];
        v.x *= s; v.y *= s; v.z *= s; v.w *= s;
        ((float4*)h)[gid] = v;
    }
}

// ---------------------------------------------------------------------------
// Kernel 2: scatter-add  h_in[dst] += x[src]  (GIN sum aggregation)
// 64 float4-chunks per edge; consecutive threads cover consecutive chunks of
// the same edge -> coalesced gather from x[src], coalesced atomics to h[dst].
// x (10.24 MB) and h (10.24 MB) are L2-resident on a 192 MB L2.
// ---------------------------------------------------------------------------
__global__ void scatter_add(const float* __restrict__ x, const int* __restrict__ ei,
                            float* __restrict__ h) {
    unsigned gid = blockIdx.x * blockDim.x + threadIdx.x;
    unsigned e = gid >> 6;          // edge id
    unsigned c = gid & 63u;         // float4 chunk within the 256-dim feature
    if (e < N_EDGES) {
        int src = ei[e];
        int dst = ei[N_EDGES + e];
        float4 v = ((const float4*)(x + (size_t)src * DIM))[c];
        float* hd = h + (size_t)dst * DIM + c * 4;
        atomicAdd(hd + 0, v.x);
        atomicAdd(hd + 1, v.y);
        atomicAdd(hd + 2, v.z);
        atomicAdd(hd + 3, v.w);
    }
}

// ---------------------------------------------------------------------------
// Kernels 3/4: fp32 GEMM via V_WMMA_F32_16X16X4_F32.
// Out[m][n] = act( sum_k A[m][k] * W[k][n] + bias[n] )
// One wave per 16x16 output tile; 8 waves (256 threads) per block.
// M = 10000 = 625*16, N = 256 = 16*16, K = 256 = 64*4  (exact tiling,
// EXEC stays all-ones as WMMA requires).
//
// Per-lane fragment layout (ISA 7.12.2, 32-bit):
//   A 16x4 (2 VGPRs): lane<16 holds (A[r][k], A[r][k+1]); lane>=16 holds
//                     (A[r][k+2], A[r][k+3])  -> one contiguous float2 load.
//   B 4x16 (2 VGPRs): v0 = row K (+2 for upper lanes), v1 = row K+1 (+2).
//   C/D 16x16 (8 VGPRs): vgpr i -> row half*8 + i, col = lane&15.
// ---------------------------------------------------------------------------
template<int RELU>
__global__ void gemm_wmma_f32(const float* __restrict__ A, const float* __restrict__ W,
                              const float* __restrict__ bias, float* __restrict__ Out) {
    int wave = threadIdx.x >> 5;               // 0..7
    int lane = threadIdx.x & 31;
    int tile = blockIdx.x * 8 + wave;          // 0..9999
    int tm   = tile >> 4;                      // 0..624  (row tile)
    int tn   = tile & 15;                      // 0..15   (col tile)
    int half = lane >> 4;                      // 0 | 1
    int l15  = lane & 15;

    int arow = tm * 16 + l15;
    int col  = tn * 16 + l15;
    const float* Ap = A + (size_t)arow * DIM;  // A row base for this lane
    const float* Wp = W + col;                 // W column base for this lane

    v8f acc = {};
#pragma unroll 8
    for (int k = 0; k < DIM; k += 4) {
        int kk = k + 2 * half;
        v2f a = *(const v2f*)(Ap + kk);        // contiguous b64 load
        v2f b;
        b.x = Wp[(size_t)kk * DIM];
        b.y = Wp[(size_t)(kk + 1) * DIM];
        acc = __builtin_amdgcn_wmma_f32_16x16x4_f32(
            /*neg_a=*/false, a, /*neg_b=*/false, b,
            /*c_mod=*/(short)0, acc, /*reuse_a=*/false, /*reuse_b=*/false);
    }

    float bv = bias[col];
    int rbase = tm * 16 + half * 8;
#pragma unroll
    for (int i = 0; i < 8; ++i) {
        float v = acc[i] + bv;
        if (RELU) v = v > 0.0f ? v : 0.0f;
        Out[(size_t)(rbase + i) * DIM + col] = v;
    }
}

// ---------------------------------------------------------------------------
// Launch: init -> scatter -> GEMM1(+ReLU) -> GEMM2
// ---------------------------------------------------------------------------
extern "C" void kernel_launch(void* const* d_in, const int* in_sizes, int n_in,
                              void* d_out, int out_size, void* d_ws, size_t ws_size,
                              hipStream_t stream) {
    const float* x   = (const float*)d_in[0];
    const int*   ei  = (const int*)  d_in[1];   // [2, E] int32 (src row, dst row)
    const float* w1  = (const float*)d_in[2];   // [256,256] row-major
    const float* b1  = (const float*)d_in[3];
    const float* w2  = (const float*)d_in[4];
    const float* b2  = (const float*)d_in[5];
    const float* eps = (const float*)d_in[6];

    float* h_in = (float*)d_ws;                          // N*DIM f32 (10.24 MB)
    float* h1   = h_in + (size_t)N_NODES * DIM;          // N*DIM f32 (10.24 MB)
    float* out  = (float*)d_out;

    // 1) h_in = (1+eps) * x
    init_h<<<(N_NODES * (DIM / 4)) / 256, 256, 0, stream>>>(x, eps, h_in);

    // 2) h_in[dst] += x[src] over all edges (E*64 float4 chunks)
    scatter_add<<<(N_EDGES * 64) / 256, 256, 0, stream>>>(x, ei, h_in);

    // 3) h1 = relu(h_in @ w1 + b1)   : 10000 tiles / 8 waves = 1250 blocks
    gemm_wmma_f32<1><<<1250, 256, 0, stream>>>(h_in, w1, b1, h1);

    // 4) out = h1 @ w2 + b2
    gemm_wmma_f32<0><<<1250, 256, 0, stream>>>(h1, w2, b2, out);
}